// ContextAwareModel_9655086481739
// MI455X (gfx1250) — compile-verified
//
#include <hip/hip_runtime.h>
#include <hip/hip_bf16.h>

typedef float v2f __attribute__((ext_vector_type(2)));
typedef float v8f __attribute__((ext_vector_type(8)));

#define VOCAB 100000
#define EMB   300
#define HD    256
#define BSZ   64
#define SEQ   512

#define KP0   278            // (EMB + HD)/2   = 556/2
#define KP1   384            // (2HD + HD)/2   = 768/2
#define NBLK  64
#define NTHR  256

// ---- workspace layout (in floats) ----
// [0,64)                      : grid barrier (cnt, gen)
// h ping-pong: 2 copies of [2 layers][2 dirs][64][256]
#define WS_H   64
#define WS_C   (WS_H + 131072)          // c: [2][2][64][256]
#define WS_B0  (WS_C + 65536)           // bias0: 2*1024
#define WS_B1  (WS_B0 + 2048)           // bias1: 2*1024
#define WS_W0  (WS_B1 + 2048)           // wpk l0: 2*278*1024 float2 = 1138688 floats
#define WS_W1  (WS_W0 + 1138688)        // wpk l1: 2*384*1024 float2 = 1572864 floats
#define WS_END (WS_W1 + 1572864)        // ~11.65 MB total

// ---------------- prep: transpose + K-pair-pack weights, fuse biases ----------------
__global__ void lstm_prep_kernel(const float* __restrict__ wih0, const float* __restrict__ whh0,
                                 const float* __restrict__ bih0, const float* __restrict__ bhh0,
                                 const float* __restrict__ wih1, const float* __restrict__ whh1,
                                 const float* __restrict__ bih1, const float* __restrict__ bhh1,
                                 float* __restrict__ ws)
{
    float2* w0 = (float2*)(ws + WS_W0);
    float2* w1 = (float2*)(ws + WS_W1);
    const int n0 = 2 * KP0 * 1024;
    const int n1 = 2 * KP1 * 1024;
    const int ntot = n0 + n1 + 4096;
    for (int i = blockIdx.x * blockDim.x + threadIdx.x; i < ntot; i += gridDim.x * blockDim.x) {
        if (i < n0) {
            int j  = i & 1023;
            int kp = (i >> 10) % KP0;
            int d  = i / (KP0 * 1024);
            int ka = 2 * kp, kb = 2 * kp + 1;
            float xa = (ka < EMB) ? wih0[(d * 1024 + j) * EMB + ka]
                                  : whh0[(d * 1024 + j) * HD + (ka - EMB)];
            float xb = (kb < EMB) ? wih0[(d * 1024 + j) * EMB + kb]
                                  : whh0[(d * 1024 + j) * HD + (kb - EMB)];
            w0[i] = make_float2(xa, xb);
        } else if (i < n0 + n1) {
            int t  = i - n0;
            int j  = t & 1023;
            int kp = (t >> 10) % KP1;
            int d  = t / (KP1 * 1024);
            int ka = 2 * kp, kb = 2 * kp + 1;
            float xa = (ka < 512) ? wih1[(d * 1024 + j) * 512 + ka]
                                  : whh1[(d * 1024 + j) * HD + (ka - 512)];
            float xb = (kb < 512) ? wih1[(d * 1024 + j) * 512 + kb]
                                  : whh1[(d * 1024 + j) * HD + (kb - 512)];
            w1[t] = make_float2(xa, xb);
        } else {
            int t = i - n0 - n1;
            if (t < 2048) ws[WS_B0 + t] = bih0[t] + bhh0[t];
            else          { int u = t - 2048; ws[WS_B1 + u] = bih1[u] + bhh1[u]; }
        }
    }
}

// ---------------- persistent LSTM ----------------
__device__ __forceinline__ float sigmoidf_(float x) { return 1.0f / (1.0f + __expf(-x)); }

__device__ __forceinline__ void grid_barrier(unsigned* bar, int nb)
{
    __threadfence();
    __syncthreads();
    if (threadIdx.x == 0) {
        unsigned g = __hip_atomic_load(&bar[1], __ATOMIC_RELAXED, __HIP_MEMORY_SCOPE_AGENT);
        unsigned a = __hip_atomic_fetch_add(&bar[0], 1u, __ATOMIC_ACQ_REL, __HIP_MEMORY_SCOPE_AGENT);
        if (a == (unsigned)(nb - 1)) {
            __hip_atomic_store(&bar[0], 0u, __ATOMIC_RELAXED, __HIP_MEMORY_SCOPE_AGENT);
            __hip_atomic_fetch_add(&bar[1], 1u, __ATOMIC_ACQ_REL, __HIP_MEMORY_SCOPE_AGENT);
        } else {
            while (__hip_atomic_load(&bar[1], __ATOMIC_ACQUIRE, __HIP_MEMORY_SCOPE_AGENT) == g) {
                __builtin_amdgcn_s_sleep(2);
            }
        }
    }
    __syncthreads();
}

// K-loop: nchunks steps of K=4 fp32 WMMA. A advances 2 v2f/chunk, B advances 2048 v2f/chunk
// (two rows of 1024 packed columns). Two accumulator chains for ILP.
__device__ __forceinline__ void kloop(const v2f* __restrict__ A2, int ai,
                                      const v2f* __restrict__ B2, int bi,
                                      int nchunks, v8f& accE, v8f& accO)
{
#pragma unroll 4
    for (int kc = 0; kc < nchunks; ++kc) {
        v2f a = A2[ai + 2 * kc];
        v2f b = B2[bi + 2048 * kc];
        if (kc & 1)
            accO = __builtin_amdgcn_wmma_f32_16x16x4_f32(false, a, false, b, (short)0, accO, false, false);
        else
            accE = __builtin_amdgcn_wmma_f32_16x16x4_f32(false, a, false, b, (short)0, accE, false, false);
    }
}

__global__ void __launch_bounds__(NTHR, 1)
lstm_persistent_kernel(const int* __restrict__ contexts, const int* __restrict__ positions,
                       const float* __restrict__ emb,
                       const float* __restrict__ wcls, const float* __restrict__ bcls,
                       float* __restrict__ ws, float* __restrict__ out)
{
    __shared__ float lds_g[2][4][16][16];   // [group-in-block][gate i/f/g/o][row m][col u]

    unsigned* bar   = (unsigned*)ws;
    float*    hbuf  = ws + WS_H;            // 2 ping-pong copies of [2][2][64][256]
    float*    cbuf  = ws + WS_C;
    const v2f* h2   = (const v2f*)hbuf;
    const v2f* e2   = (const v2f*)emb;
    const v2f* w0   = (const v2f*)(ws + WS_W0);
    const v2f* w1   = (const v2f*)(ws + WS_W1);

    const int tid   = threadIdx.x;
    const int lane  = tid & 31;
    const int wid   = tid >> 5;
    const int gid   = blockIdx.x * 8 + wid;      // 512 waves
    const int gate  = gid & 3;                   // i,f,g,o
    const int ggrp  = gid >> 2;                  // 128 tile groups
    const int ut    = ggrp & 15;                 // hidden-unit tile
    const int mt    = (ggrp >> 4) & 3;           // batch tile
    const int dir   = ggrp >> 6;                 // direction
    const int grpl  = wid >> 2;                  // group within block (0/1)

    const int lq    = lane & 15;
    const int lhalf = lane >> 4;
    const int n0col = gate * HD + ut * 16;       // gate-major gate column base
    const int row   = mt * 16 + lq;              // batch row for this lane's A fragment

    // B fragment base indices (v2f units): kp*1024 + j, with kp = 2*kc + lhalf added in kloop
    const int bbase0 = dir * (KP0 * 1024) + lhalf * 1024 + n0col + lq;
    const int bbase1 = dir * (KP1 * 1024) + lhalf * 1024 + n0col + lq;

    const float biasL0 = ws[WS_B0 + dir * 1024 + n0col + lq];
    const float biasL1 = ws[WS_B1 + dir * 1024 + n0col + lq];

    // A fragment bases within one h copy (v2f units): ((layer*2+d)*64+row)*128 + lhalf
    const int ah00 = ((0 * 2 + 0)  * 64 + row) * 128 + lhalf;
    const int ah01 = ((0 * 2 + 1)  * 64 + row) * 128 + lhalf;
    const int ah0d = ((0 * 2 + dir) * 64 + row) * 128 + lhalf;
    const int ah1d = ((1 * 2 + dir) * 64 + row) * 128 + lhalf;

    // cell-update ownership (fixed across t -> c needs no double buffer)
    const int dirB = (blockIdx.x * 2) >> 6;
    const int mtB  = ((blockIdx.x * 2) >> 4) & 3;
    const int m_u  = tid >> 4;
    const int uc_u = tid & 15;
    const int bglob = mtB * 16 + m_u;
    const int myPos = positions[bglob];

    for (int t = 0; t < SEQ; ++t) {
        const int hprev = ((t + 1) & 1) * 65536 / 2;   // v2f offset of h(t-1) copy
        const int hcurr = (t & 1) * 65536 / 2;         // v2f offset of h(t) copy
        const int hwofs = (t & 1) * 65536;             // float offset for h writes

        // ---------------- stage A: layer 0 gates ----------------
        int tok = contexts[row * SEQ + t];
        v8f accE, accO;
#pragma unroll
        for (int r = 0; r < 8; ++r) { accE[r] = biasL0; accO[r] = 0.0f; }
        kloop(e2, tok * 150 + lhalf, w0, bbase0,              75, accE, accO); // x @ Wih
        kloop(h2, hprev + ah0d,      w0, bbase0 + 150 * 1024, 64, accE, accO); // h0 @ Whh
        v8f acc = accE + accO;
#pragma unroll
        for (int r = 0; r < 8; ++r)
            lds_g[grpl][gate][r + 8 * lhalf][lq] = acc[r];
        __syncthreads();

        // layer-0 cell update (each thread owns 2 elements)
#pragma unroll
        for (int grp = 0; grp < 2; ++grp) {
            int uglob = (((blockIdx.x * 2 + grp) & 15) * 16) + uc_u;
            float iv = lds_g[grp][0][m_u][uc_u];
            float fv = lds_g[grp][1][m_u][uc_u];
            float gv = lds_g[grp][2][m_u][uc_u];
            float ov = lds_g[grp][3][m_u][uc_u];
            int cidx = ((0 * 2 + dirB) * 64 + bglob) * 256 + uglob;
            float c_old = cbuf[cidx];
            iv = sigmoidf_(iv); fv = sigmoidf_(fv); ov = sigmoidf_(ov); gv = tanhf(gv);
            float cn = fv * c_old + iv * gv;
            float hn = ov * tanhf(cn);
            cbuf[cidx] = cn;
            hbuf[hwofs + cidx] = hn;
        }
        grid_barrier(bar, NBLK);

        // ---------------- stage B: layer 1 gates ----------------
#pragma unroll
        for (int r = 0; r < 8; ++r) { accE[r] = biasL1; accO[r] = 0.0f; }
        kloop(h2, hcurr + ah00, w1, bbase1,              64, accE, accO); // x1[:,0:256]
        kloop(h2, hcurr + ah01, w1, bbase1 + 128 * 1024, 64, accE, accO); // x1[:,256:512]
        kloop(h2, hprev + ah1d, w1, bbase1 + 256 * 1024, 64, accE, accO); // h1 @ Whh
        acc = accE + accO;
#pragma unroll
        for (int r = 0; r < 8; ++r)
            lds_g[grpl][gate][r + 8 * lhalf][lq] = acc[r];
        __syncthreads();

        // layer-1 cell update + scatter of encoded state at target position
#pragma unroll
        for (int grp = 0; grp < 2; ++grp) {
            int uglob = (((blockIdx.x * 2 + grp) & 15) * 16) + uc_u;
            float iv = lds_g[grp][0][m_u][uc_u];
            float fv = lds_g[grp][1][m_u][uc_u];
            float gv = lds_g[grp][2][m_u][uc_u];
            float ov = lds_g[grp][3][m_u][uc_u];
            int cidx = ((1 * 2 + dirB) * 64 + bglob) * 256 + uglob;
            float c_old = cbuf[cidx];
            iv = sigmoidf_(iv); fv = sigmoidf_(fv); ov = sigmoidf_(ov); gv = tanhf(gv);
            float cn = fv * c_old + iv * gv;
            float hn = ov * tanhf(cn);
            cbuf[cidx] = cn;
            hbuf[hwofs + cidx] = hn;
            if (myPos == t)
                out[256 + bglob * 512 + dirB * 256 + uglob] = hn;  // target_output
        }
        grid_barrier(bar, NBLK);
    }

    // ---------------- classifier head: logits + probs ----------------
    if (blockIdx.x == 0 && tid < 128) {
        int b = tid >> 1, k = tid & 1;
        float s = bcls[k];
        const float* tp = out + 256 + b * 512;
        const float* wp = wcls + k * 512;
        for (int j = 0; j < 512; ++j) s += tp[j] * wp[j];
        out[b * 2 + k]       = s;               // logits
        out[128 + b * 2 + k] = sigmoidf_(s);    // probs
    }
}

extern "C" void kernel_launch(void* const* d_in, const int* in_sizes, int n_in,
                              void* d_out, int out_size, void* d_ws, size_t ws_size,
                              hipStream_t stream)
{
    (void)in_sizes; (void)n_in; (void)out_size; (void)ws_size;
    const int*   contexts = (const int*)d_in[0];
    const int*   positions= (const int*)d_in[1];
    const float* emb  = (const float*)d_in[2];
    const float* wih0 = (const float*)d_in[3];
    const float* whh0 = (const float*)d_in[4];
    const float* bih0 = (const float*)d_in[5];
    const float* bhh0 = (const float*)d_in[6];
    const float* wih1 = (const float*)d_in[7];
    const float* whh1 = (const float*)d_in[8];
    const float* bih1 = (const float*)d_in[9];
    const float* bhh1 = (const float*)d_in[10];
    const float* wcls = (const float*)d_in[11];
    const float* bcls = (const float*)d_in[12];
    float* out = (float*)d_out;
    float* ws  = (float*)d_ws;

    // zero barrier + h (both ping-pong copies) + c each call (deterministic state)
    hipMemsetAsync(ws, 0, (size_t)(64 + 131072 + 65536) * sizeof(float), stream);
    lstm_prep_kernel<<<512, 256, 0, stream>>>(wih0, whh0, bih0, bhh0,
                                              wih1, whh1, bih1, bhh1, ws);
    lstm_persistent_kernel<<<NBLK, NTHR, 0, stream>>>(contexts, positions, emb,
                                                      wcls, bcls, ws, out);
}